// CAM_39737037423022
// MI455X (gfx1250) — compile-verified
//
#include <hip/hip_runtime.h>

#define Dg 64
#define Cch 256
#define Rh 128

typedef __attribute__((ext_vector_type(2))) float v2f;
typedef __attribute__((ext_vector_type(8))) float v8f;

// ---------------- index grid build ----------------
__global__ void init_idx(int* __restrict__ idxGrid) {
    idxGrid[blockIdx.x * 256 + threadIdx.x] = -1;   // 1024 blocks -> 64^3 ints
}

__global__ void scatter_idx(const int* __restrict__ coords, int* __restrict__ idxGrid, int n) {
    int i = blockIdx.x * 256 + threadIdx.x;
    if (i < n) {
        int ix = coords[3 * i + 0];
        int iy = coords[3 * i + 1];
        int iz = coords[3 * i + 2];
        idxGrid[(ix * Dg + iy) * Dg + iz] = i;
    }
}

// ---------------- pass 1: pool along x, reading feats through idx grid ------
// block = (y,z), thread = channel c. Each thread owns the whole x-line (64) in regs.
__global__ __launch_bounds__(256) void pool_x_gather(const int* __restrict__ idxGrid,
                                                     const float* __restrict__ feats,
                                                     float* __restrict__ out) {
    const int b = blockIdx.x;          // y*64 + z
    const int y = b >> 6, z = b & 63;
    const int c = threadIdx.x;
    const int yz = y * Dg + z;
    const float NEG = -__builtin_inff();

    float v[Dg];
#pragma unroll
    for (int x = 0; x < Dg; ++x) {
        int idx = idxGrid[x * (Dg * Dg) + yz];      // uniform per block -> scalar-cached
        v[x] = (idx >= 0) ? feats[(long)idx * Cch + c] : NEG;
    }
    const long base = (long)yz * Cch + c;           // + x*1048576
#pragma unroll
    for (int x = 0; x < Dg; ++x) {
        float m = v[x];
#pragma unroll
        for (int d = -3; d <= 3; ++d) {
            int j = x + d;
            if (d != 0 && j >= 0 && j < Dg) m = fmaxf(m, v[j]);
        }
        out[base + (long)x * (Dg * Dg * Cch)] = m;
    }
}

// ---------------- dense 1-D max pool along a templated stride ----------------
// base = b0*S0 + b1*S1 + tid ; element i at base + i*STRIDE (floats)
template <int STRIDE, int S0, int S1>
__global__ __launch_bounds__(256) void pool_line(const float* __restrict__ in,
                                                 float* __restrict__ out) {
    const int b = blockIdx.x;
    const int b0 = b >> 6, b1 = b & 63;
    const long base = (long)b0 * S0 + (long)b1 * S1 + threadIdx.x;
    float v[Dg];
#pragma unroll
    for (int i = 0; i < Dg; ++i) v[i] = in[base + (long)i * STRIDE];
#pragma unroll
    for (int i = 0; i < Dg; ++i) {
        float m = v[i];
#pragma unroll
        for (int d = -3; d <= 3; ++d) {
            int j = i + d;
            if (d != 0 && j >= 0 && j < Dg) m = fmaxf(m, v[j]);
        }
        out[base + (long)i * STRIDE] = m;
    }
}

// ---------------- fused gather + MLP (f32 WMMA) ------------------------------
// One block (8 wave32s) handles 16 points. GEMM1: wave w -> hidden cols [16w,16w+16).
// GEMM2: wave w -> output cols [32w, 32w+32) as two 16x16 tiles.
__global__ __launch_bounds__(256) void mlp_kernel(const float* __restrict__ pooled,  // buf0
                                                  const float* __restrict__ feats,
                                                  const int* __restrict__ coords,
                                                  const float* __restrict__ W1,
                                                  const float* __restrict__ W2,
                                                  float* __restrict__ out, int n) {
    __shared__ __align__(16) float yT[16][258];   // pooled features (padded rows)
    __shared__ __align__(16) float fT[16][258];   // raw features
    __shared__ __align__(16) float hT[16][132];   // hidden activations

    const int t    = threadIdx.x;
    const int base = blockIdx.x * 16;
    const int wave = t >> 5;
    const int lane = t & 31;
    const int m    = lane & 15;      // output row within tile
    const int kh   = lane >> 4;      // K-half selector for A/B fragments

    // ---- cooperative gather: 16 coalesced 1 KB rows each for pooled + feats
    for (int r = 0; r < 16; ++r) {
        int p = base + r; if (p >= n) p = n - 1;          // clamp (keeps EXEC full)
        int ix = coords[3 * p + 0];
        int iy = coords[3 * p + 1];
        int iz = coords[3 * p + 2];
        long off = (((long)ix * Dg + iy) * Dg + iz) * Cch;
        yT[r][t] = pooled[off + t];
        fT[r][t] = feats[(long)p * Cch + t];
    }
    __syncthreads();

    // ---- GEMM1: H(16x128) = relu(Y(16x256) @ W1(256x128)), K-step 4
    {
        const int col = wave * 16 + m;
        v8f acc = {};
        for (int k0 = 0; k0 < Cch; k0 += 4) {
            int k = k0 + 2 * kh;
            v2f a = *(const v2f*)&yT[m][k];                 // A: M=m, K=k,k+1
            v2f b;
            b.x = W1[(long)k * Rh + col];                   // B: K=k,   N=col
            b.y = W1[(long)(k + 1) * Rh + col];             //    K=k+1, N=col
            acc = __builtin_amdgcn_wmma_f32_16x16x4_f32(false, a, false, b,
                                                        (short)0, acc, false, false);
        }
#pragma unroll
        for (int r = 0; r < 8; ++r) {
            int row = r + 8 * kh;                           // C/D layout
            hT[row][wave * 16 + m] = fmaxf(acc[r], 0.0f);   // ReLU
        }
    }
    __syncthreads();

    // ---- GEMM2: Z(16x256) = H(16x128) @ W2(128x256); sigmoid; *feats; store
#pragma unroll
    for (int tt = 0; tt < 2; ++tt) {
        const int col = wave * 32 + tt * 16 + m;
        v8f acc = {};
        for (int k0 = 0; k0 < Rh; k0 += 4) {
            int k = k0 + 2 * kh;
            v2f a = *(const v2f*)&hT[m][k];
            v2f b;
            b.x = W2[(long)k * Cch + col];
            b.y = W2[(long)(k + 1) * Cch + col];
            acc = __builtin_amdgcn_wmma_f32_16x16x4_f32(false, a, false, b,
                                                        (short)0, acc, false, false);
        }
#pragma unroll
        for (int r = 0; r < 8; ++r) {
            int row = r + 8 * kh;
            int p = base + row;
            if (p < n) {
                float s = 1.0f / (1.0f + __expf(-acc[r]));  // sigmoid
                out[(long)p * Cch + col] = fT[row][col] * s;
            }
        }
    }
}

extern "C" void kernel_launch(void* const* d_in, const int* in_sizes, int n_in,
                              void* d_out, int out_size, void* d_ws, size_t ws_size,
                              hipStream_t stream) {
    const float* feats  = (const float*)d_in[0];
    const int*   coords = (const int*)d_in[1];
    const float* W1     = (const float*)d_in[2];
    const float* W2     = (const float*)d_in[3];
    float*       outp   = (float*)d_out;
    const int N = in_sizes[0] / Cch;                        // 100000

    // workspace layout: [idxGrid 1MB][buf0 256MB][buf1 256MB]
    char*  ws      = (char*)d_ws;
    int*   idxGrid = (int*)ws;
    float* buf0    = (float*)(ws + (1u << 20));
    float* buf1    = buf0 + (size_t)Dg * Dg * Dg * Cch;

    init_idx<<<(Dg * Dg * Dg) / 256, 256, 0, stream>>>(idxGrid);
    scatter_idx<<<(N + 255) / 256, 256, 0, stream>>>(coords, idxGrid, N);

    // pass 1: pool over x (indirect through idx grid) -> buf0
    pool_x_gather<<<Dg * Dg, 256, 0, stream>>>(idxGrid, feats, buf0);
    // pass 2: pool over y (line stride 64*256), blocks over (x,z)
    pool_line<Dg * Cch, Dg * Dg * Cch, Cch><<<Dg * Dg, 256, 0, stream>>>(buf0, buf1);
    // pass 3: pool over z (line stride 256), blocks over (x,y)
    pool_line<Cch, Dg * Dg * Cch, Dg * Cch><<<Dg * Dg, 256, 0, stream>>>(buf1, buf0);

    // fused gather + MLP + elementwise
    mlp_kernel<<<(N + 15) / 16, 256, 0, stream>>>(buf0, feats, coords, W1, W2, outp, N);
}